// MessagePassing_7164005450127
// MI455X (gfx1250) — compile-verified
//
#include <hip/hip_runtime.h>

// Graph message passing: gather along src, scatter-sum + scatter-product at dst.
// Memory/atomic-bound (no GEMM structure -> WMMA inapplicable); all hot data is
// L2-resident on MI455X (192 MB), so per-element f32 atomics at L2 are the
// fastest correct path.

#define N_NODES 50000
#define N_EDGES 800000
#define D_FEAT  64

// ---------------------------------------------------------------------------
// Init: out_sum = 0, out_prod = 1  (vectorized float4 stores)
// ---------------------------------------------------------------------------
__global__ __launch_bounds__(256) void mp_init_kernel(float4* __restrict__ out_sum4,
                                                      float4* __restrict__ out_prod4)
{
    const int total = (N_NODES * D_FEAT) / 4;  // 800,000 float4
    int i = blockIdx.x * blockDim.x + threadIdx.x;
    if (i < total) {
        out_sum4[i]  = make_float4(0.0f, 0.0f, 0.0f, 0.0f);
        out_prod4[i] = make_float4(1.0f, 1.0f, 1.0f, 1.0f);
    }
}

// ---------------------------------------------------------------------------
// CAS-loop float multiply. Seeded with bit-pattern of 1.0f (the init value),
// so the first multiply on each address usually succeeds with zero loads; on
// failure atomicCAS returns the live value and we retry.
// ---------------------------------------------------------------------------
__device__ __forceinline__ void atomic_mul_f32(float* addr, float v)
{
    unsigned int* ua = reinterpret_cast<unsigned int*>(addr);
    unsigned int old = 0x3F800000u;  // guess: 1.0f (init value)
    unsigned int assumed;
    do {
        assumed = old;
        float cur = __uint_as_float(assumed);
        old = atomicCAS(ua, assumed, __float_as_uint(cur * v));
    } while (old != assumed);
}

// ---------------------------------------------------------------------------
// Scatter kernel: 16 lanes per edge, each lane owns a 4-float (16 B) chunk of
// the 64-float feature row -> b128 gathers, 4x atomic f32 add, 4x CAS multiply.
// Edge indices are streamed once -> non-temporal loads (don't pollute caches
// holding the reused feature/output rows).
// ---------------------------------------------------------------------------
__global__ __launch_bounds__(256) void mp_scatter_kernel(
    const float* __restrict__ x_sum,
    const float* __restrict__ x_prod,
    const long long* __restrict__ src_idx,
    const long long* __restrict__ dst_idx,
    float* __restrict__ out_sum,
    float* __restrict__ out_prod)
{
    unsigned int tid = blockIdx.x * blockDim.x + threadIdx.x;
    unsigned int e = tid >> 4;            // edge id: 16 threads per edge
    unsigned int c = (tid & 15u) << 2;    // feature chunk start: 0,4,...,60
    if (e >= N_EDGES) return;

    // int64 node ids; values < 50000 so 32-bit offset math is exact.
    const int src = (int)__builtin_nontemporal_load(src_idx + e);
    const int dst = (int)__builtin_nontemporal_load(dst_idx + e);

    const long long soff = (long long)src * D_FEAT + c;
    const long long doff = (long long)dst * D_FEAT + c;

    // 128-bit gathers (L2-resident feature matrices)
    const float4 ms = *reinterpret_cast<const float4*>(x_sum  + soff);
    const float4 mp = *reinterpret_cast<const float4*>(x_prod + soff);

    float* os = out_sum  + doff;
    float* op = out_prod + doff;

    // scatter-sum: native f32 atomic adds (resolve at L2)
    atomicAdd(os + 0, ms.x);
    atomicAdd(os + 1, ms.y);
    atomicAdd(os + 2, ms.z);
    atomicAdd(os + 3, ms.w);

    // scatter-product: CAS multiply (no HW atomic mul exists)
    atomic_mul_f32(op + 0, mp.x);
    atomic_mul_f32(op + 1, mp.y);
    atomic_mul_f32(op + 2, mp.z);
    atomic_mul_f32(op + 3, mp.w);
}

// ---------------------------------------------------------------------------
// Launch
// ---------------------------------------------------------------------------
extern "C" void kernel_launch(void* const* d_in, const int* in_sizes, int n_in,
                              void* d_out, int out_size, void* d_ws, size_t ws_size,
                              hipStream_t stream)
{
    (void)in_sizes; (void)n_in; (void)out_size; (void)d_ws; (void)ws_size;

    const float*     x_sum  = (const float*)d_in[0];               // [50000, 64] f32
    const float*     x_prod = (const float*)d_in[1];               // [50000, 64] f32
    const long long* eidx   = (const long long*)d_in[2];           // [2, 800000] i64
    const long long* src    = eidx;                                // row 0
    const long long* dst    = eidx + N_EDGES;                      // row 1

    float* out_sum  = (float*)d_out;                               // [50000, 64]
    float* out_prod = (float*)d_out + (size_t)N_NODES * D_FEAT;    // [50000, 64]

    // 1) init outputs (0 / 1)
    {
        const int total4 = (N_NODES * D_FEAT) / 4;                 // 800,000
        const int block = 256;
        const int grid  = (total4 + block - 1) / block;
        mp_init_kernel<<<grid, block, 0, stream>>>((float4*)out_sum, (float4*)out_prod);
    }

    // 2) scatter: 16 threads/edge -> 12.8M threads
    {
        const long long total = (long long)N_EDGES * 16;           // 12,800,000
        const int block = 256;
        const int grid  = (int)((total + block - 1) / block);      // 50,000 blocks
        mp_scatter_kernel<<<grid, block, 0, stream>>>(x_sum, x_prod, src, dst,
                                                      out_sum, out_prod);
    }
}